// QuantumLayer_77781857731051
// MI455X (gfx1250) — compile-verified
//
#include <hip/hip_runtime.h>
#include <math.h>

// QuantumLayer (8-qubit RY-encode + RX layer + CNOT ring + <Z_i>) collapsed to
// closed form:
//   theta_j = (x_j - mean)/(std_ddof1 + 1e-6) * pi
//   z_j     = cos(w_j) * cos(theta_j)
//   out[i]  = z_0*...*z_i          (i = 1..7)   [prefix products]
//   out[0]  = z_1*...*z_7
//
// Memory-bound streaming kernel: one thread per sample, 65536 threads = 2048
// wave32 waves. Total HBM traffic 4 MB -> ~0.18us at 23.3 TB/s; launch-bound.
// No matrix op survives the collapse, so WMMA is deliberately absent (forcing
// `probs @ signs` through v_wmma would need a 65536x256 intermediate = 64 MB,
// >10x worse on the roofline).
//
// cos is computed with the raw hardware transcendental v_cos_f32
// (__builtin_amdgcn_cosf takes its input in revolutions = radians/2pi).
// Inputs are provably in-range: standardized values with ddof=1 over 8
// elements are bounded by 7/sqrt(8) ~= 2.475, so |theta_rev| <= 1.24, and the
// shared weights are tiny. This avoids inlining OCML's Payne-Hanek slow path
// 16x per thread (which was ~2500 extra SALU/VALU instructions).

__global__ __launch_bounds__(256) void qlayer_kernel(
    const float* __restrict__ x,     // (n, 8)
    const float* __restrict__ w,     // (1, 8), uniform across threads
    float* __restrict__ out,         // (n, 8)
    int n)
{
    int b = blockIdx.x * blockDim.x + threadIdx.x;
    if (b >= n) return;

    // CDNA5 prefetch path (global_prefetch_b8); stream ~128KB ahead.
    int pf = b + 4096;
    if (pf < n) __builtin_prefetch(x + (size_t)pf * 8, 0, 1);

    const float INV_2PI = 0.15915494309189535f;

    // Uniform weights -> scalar loads; cos(w_j) shared by every sample.
    float cw[8];
#pragma unroll
    for (int j = 0; j < 8; ++j)
        cw[j] = __builtin_amdgcn_cosf(w[j] * INV_2PI);

    // 32-byte row, 32-byte aligned -> two global_load_b128.
    const float4* xv = (const float4*)(x + (size_t)b * 8);
    float4 lo = xv[0];
    float4 hi = xv[1];
    float xs[8] = {lo.x, lo.y, lo.z, lo.w, hi.x, hi.y, hi.z, hi.w};

    // Per-sample standardization, unbiased std (ddof = 1 over 8 features).
    float mean = 0.0f;
#pragma unroll
    for (int j = 0; j < 8; ++j) mean += xs[j];
    mean *= 0.125f;

    float var = 0.0f;
#pragma unroll
    for (int j = 0; j < 8; ++j) {
        float d = xs[j] - mean;
        var += d * d;
    }
    float stdv = sqrtf(var * (1.0f / 7.0f));

    // theta_j in revolutions: (x_j - mean)/(std+1e-6) * pi / (2*pi)
    //                       = (x_j - mean) * 0.5/(std+1e-6)
    float srev = 0.5f / (stdv + 1e-6f);

    float z[8];
#pragma unroll
    for (int j = 0; j < 8; ++j) {
        float trev = (xs[j] - mean) * srev;
        z[j] = cw[j] * __builtin_amdgcn_cosf(trev);
    }

    // Prefix products for wires 1..7; wire 0 gets z1*...*z7 (the CNOT-ring
    // wraparound XORs bit 0 out of the final permutation's bit-0 parity set).
    float o[8];
    float pre = z[0];
#pragma unroll
    for (int i = 1; i < 8; ++i) {
        pre *= z[i];
        o[i] = pre;           // z0*...*zi
    }
    float suf = z[1];
#pragma unroll
    for (int i = 2; i < 8; ++i) suf *= z[i];
    o[0] = suf;               // z1*...*z7

    // Two global_store_b128.
    float4* ov = (float4*)(out + (size_t)b * 8);
    ov[0] = make_float4(o[0], o[1], o[2], o[3]);
    ov[1] = make_float4(o[4], o[5], o[6], o[7]);
}

extern "C" void kernel_launch(void* const* d_in, const int* in_sizes, int n_in,
                              void* d_out, int out_size, void* d_ws, size_t ws_size,
                              hipStream_t stream) {
    const float* x = (const float*)d_in[0];   // (65536, 8) f32
    const float* w = (const float*)d_in[1];   // (1, 8) f32
    float* out = (float*)d_out;               // (65536, 8) f32

    int n = in_sizes[0] / 8;
    const int block = 256;                    // 8 wave32 waves per block
    int grid = (n + block - 1) / block;
    qlayer_kernel<<<grid, block, 0, stream>>>(x, w, out, n);
}